// OUT_36867999268870
// MI455X (gfx1250) — compile-verified
//
#include <hip/hip_runtime.h>

// Shapes from the reference: x (32,12,325,64) f32, W1 (64,128), b1 (64,), W2 (1,64), b2 (1,)
// out (32,12,325,325) f32 = 40,560,000 elements.
//
// Roofline: output write 162.2 MB dominates (x read 31.9 MB); ~8.3 us floor at 23.3 TB/s.
// Phase 2 (a/b dot products) is mapped onto V_WMMA_F32_16X16X4_F32.

typedef __attribute__((ext_vector_type(2))) float v2f;
typedef __attribute__((ext_vector_type(8))) float v8f;

#define BT     384              // B*T
#define NTOK   325
#define DDIM   64
#define NROWS  (BT * NTOK)      // 124800 == 16 * 7800 exactly
#define AB_OFF 256              // float offset of the a/b arrays inside d_ws
#define TILES_PER_BT 13         // 325 = 13 * 25
#define ROWS_PER_BLK 25

// ---------------------------------------------------------------------------
// Kernel 1: va[d] = sum_i W1[i, d] * w2[i]; vb[d] = sum_i W1[i, 64+d] * w2[i];
//           c = b1 . w2 + b2.   ws layout: [0..63]=va [64..127]=vb [128]=c.
// ---------------------------------------------------------------------------
__global__ __launch_bounds__(64) void prep_kernel(const float* __restrict__ W1,
                                                  const float* __restrict__ b1,
                                                  const float* __restrict__ W2,
                                                  const float* __restrict__ b2,
                                                  float* __restrict__ ws) {
    int d = threadIdx.x;  // 0..63
    float sva = 0.f, svb = 0.f;
#pragma unroll 8
    for (int i = 0; i < DDIM; ++i) {
        float w = W2[i];
        sva += W1[i * 2 * DDIM + d] * w;
        svb += W1[i * 2 * DDIM + DDIM + d] * w;
    }
    ws[d] = sva;
    ws[DDIM + d] = svb;
    if (d == 0) {
        float c = b2[0];
        for (int i = 0; i < DDIM; ++i) c += b1[i] * W2[i];
        ws[128] = c;
    }
}

// ---------------------------------------------------------------------------
// Kernel 2: a[row] = x[row,:].va ; b[row] = x[row,:].vb  via WMMA f32 16x16x4.
// One wave per 16 rows; GEMM is (16 x 64) * (64 x 16) with only N=0 (va) and
// N=1 (vb) columns non-zero. 7800 waves total (exact).
// ---------------------------------------------------------------------------
__global__ __launch_bounds__(256) void ab_kernel(const float* __restrict__ x,
                                                 float* __restrict__ ws) {
    __shared__ float sB[DDIM * 16];           // B operand, sB[k*16 + n]
    const float* va = ws;
    const float* vb = ws + DDIM;
    float* aout = ws + AB_OFF;
    float* bout = ws + AB_OFF + NROWS;

    for (int i = threadIdx.x; i < DDIM * 16; i += 256) {
        int k = i >> 4, n = i & 15;
        sB[i] = (n == 0) ? va[k] : (n == 1) ? vb[k] : 0.f;
    }
    __syncthreads();

    int lane  = threadIdx.x & 31;
    int wave  = blockIdx.x * (256 / 32) + (threadIdx.x >> 5);
    int row0  = wave * 16;
    int m     = lane & 15;                    // row within tile / column index n
    int khalf = (lane >> 4) << 1;             // 0 (lanes 0-15) or 2 (lanes 16-31)

    const float* xrow = x + (size_t)(row0 + m) * DDIM;

    v8f acc = {};
#pragma unroll
    for (int kk = 0; kk < 16; ++kk) {
        int k0 = kk * 4 + khalf;              // even -> 8B aligned
        v2f A = *(const v2f*)(xrow + k0);     // A VGPR0=K(k0), VGPR1=K(k0+1)
        v2f Bv;
        Bv[0] = sB[k0 * 16 + m];              // B row K=k0,   col n=m
        Bv[1] = sB[(k0 + 1) * 16 + m];        // B row K=k0+1, col n=m
        acc = __builtin_amdgcn_wmma_f32_16x16x4_f32(
            /*neg_a=*/false, A, /*neg_b=*/false, Bv,
            /*c_mod=*/(short)0, acc, /*reuse_a=*/false, /*reuse_b=*/false);
    }

    // D layout (32-bit C/D 16x16): lane holds column N=lane&15; VGPR r holds
    // M = r (lanes 0-15) or M = 8+r (lanes 16-31). Column 0 = a, column 1 = b.
    if (m < 2) {
        float* dst = ((m == 0) ? aout : bout) + row0 + ((lane >> 4) << 3);
#pragma unroll
        for (int r = 0; r < 8; ++r) dst[r] = acc[r];
    }
}

// ---------------------------------------------------------------------------
// Kernel 3: out[bt, n1, n2] = a[bt,n1] + b[bt,n2] + c  -- pure streaming write.
// 352 threads (11 waves) cover n2 in one shot; 25 rows per block; NT stores.
// ---------------------------------------------------------------------------
__global__ __launch_bounds__(352) void bcast_kernel(const float* __restrict__ ws,
                                                    float* __restrict__ out) {
    __shared__ float sa[ROWS_PER_BLK];
    int bt   = blockIdx.x / TILES_PER_BT;
    int tile = blockIdx.x % TILES_PER_BT;

    const float* a = ws + AB_OFF + bt * NTOK + tile * ROWS_PER_BLK;
    const float* b = ws + AB_OFF + NROWS + bt * NTOK;
    float c = ws[128];

    int t = threadIdx.x;
    if (t < ROWS_PER_BLK) sa[t] = a[t];
    __syncthreads();

    if (t < NTOK) {
        float bv = b[t] + c;
        float* o = out + (size_t)bt * NTOK * NTOK
                       + (size_t)tile * ROWS_PER_BLK * NTOK + t;
#pragma unroll
        for (int r = 0; r < ROWS_PER_BLK; ++r) {
            __builtin_nontemporal_store(sa[r] + bv, o);
            o += NTOK;
        }
    }
}

// ---------------------------------------------------------------------------
extern "C" void kernel_launch(void* const* d_in, const int* in_sizes, int n_in,
                              void* d_out, int out_size, void* d_ws, size_t ws_size,
                              hipStream_t stream) {
    (void)in_sizes; (void)n_in; (void)out_size; (void)ws_size;
    const float* x  = (const float*)d_in[0];
    const float* W1 = (const float*)d_in[1];
    const float* b1 = (const float*)d_in[2];
    const float* W2 = (const float*)d_in[3];
    const float* b2 = (const float*)d_in[4];
    float* ws  = (float*)d_ws;   // needs (256 + 2*124800)*4 bytes ~= 1.0 MB
    float* out = (float*)d_out;

    prep_kernel <<<1, 64, 0, stream>>>(W1, b1, W2, b2, ws);
    ab_kernel   <<<NROWS / 16 / 8, 256, 0, stream>>>(x, ws);        // 975 blocks
    bcast_kernel<<<BT * TILES_PER_BT, 352, 0, stream>>>(ws, out);   // 4992 blocks
}